// QFTAttentionHead_57784490000561
// MI455X (gfx1250) — compile-verified
//
#include <hip/hip_runtime.h>
#include <math.h>

// QFTAttentionHead for MI455X (gfx1250, wave32).
//
// Per token only x[:5] matters; theta is dead (|exp(i*phase)| == 1 cancels in
// |psi|^2).  Each wave handles 16 tokens: psi (32 reals) built per lane,
// psi' = QFT_re@psi / QFT_im@psi via chained V_WMMA_F32_16X16X4_F32 (fp32
// matrix pipe, K=32 as 8 chunks of 4), PauliZ expectations as a signed
// reduction over the C/D tile layout + one xor-16 lane shuffle.
//
// QFT coefficient generation is pure arithmetic (no tables/shuffles):
//   chunk kc uses columns k0 = 4kc+2lh (even), k1 = k0+1 (odd);
//   coef(m,k) = SC*e^{i*m*k*pi/16} advances per chunk by W = e^{i*m*pi/4};
//   rows m+16 obey QFT[m+16][k] = QFT[m][k]*(-1)^k -> hi tile = lo tile with
//   the odd-k (.y) component sign-flipped.

typedef __attribute__((ext_vector_type(2))) float v2f;
typedef __attribute__((ext_vector_type(4))) float v4f;
typedef __attribute__((ext_vector_type(8))) float v8f;

#define WIRES 5

__global__ __launch_bounds__(256) void qft_head_kernel(
    const float* __restrict__ x, float* __restrict__ out, int ntok)
{
    const int lane = threadIdx.x & 31;
    const int lm   = lane & 15;   // token column / QFT row within half
    const int lh   = lane >> 4;   // K-pair selector for A/B operands
    const int wid  = blockIdx.x * (blockDim.x >> 5) + (threadIdx.x >> 5);
    const int base = wid << 4;    // 16 tokens per wave
    if (base >= ntok) return;     // wave-uniform guard (EXEC stays all-ones)

    const int tok  = base + lm;
    const int tokL = tok < ntok ? tok : (ntok - 1);

    // ---- strided gather: 5 floats out of each 512-float token row --------
    // Non-temporal: only ~33MB of cachelines of the 536MB input are live.
    const float* xp = x + (size_t)tokL * 512;
    v4f   x4 = __builtin_nontemporal_load((const v4f*)xp);
    float x5 = __builtin_nontemporal_load(xp + 4);

    // Arguments are 0.5*N(0,1): tiny, so hardware sin/cos is ~1e-7 accurate.
    float c0,s0,c1,s1,c2,s2,c3,s3,c4,s4;
    __sincosf(0.5f * x4.x, &s0, &c0);
    __sincosf(0.5f * x4.y, &s1, &c1);
    __sincosf(0.5f * x4.z, &s2, &c2);
    __sincosf(0.5f * x4.w, &s3, &c3);
    __sincosf(0.5f * x5,   &s4, &c4);

    // ---- product state through wires 0..3 (16 amplitudes) ----------------
    float P4[4], P8[8], P16[16];
    P4[0] = c0*c1; P4[1] = c0*s1; P4[2] = s0*c1; P4[3] = s0*s1;
    #pragma unroll
    for (int j = 0; j < 4; ++j) { P8[2*j]  = P4[j]*c2; P8[2*j+1]  = P4[j]*s2; }
    #pragma unroll
    for (int j = 0; j < 8; ++j) { P16[2*j] = P8[j]*c3; P16[2*j+1] = P8[j]*s3; }
    // Chunk kc needs PSI[4kc+2lh + {0,1}] = P16[2kc+lh] * {c4, s4}
    float Q[8];
    #pragma unroll
    for (int kc = 0; kc < 8; ++kc) Q[kc] = lh ? P16[2*kc+1] : P16[2*kc];

    // ---- per-lane QFT coefficients with incremental rotation -------------
    const float PI16 = 0.19634954084936207f;   // pi/16
    const float SC   = 0.17677669529663689f;   // 1/sqrt(32)
    float w_re, w_im;                          // W = e^{i*lm*pi/4}
    __sincosf((float)lm * (4.0f * PI16), &w_im, &w_re);
    float r0_re, r0_im, r1_re, r1_im;          // SC*e^{i*lm*k*pi/16}, k=k0,k1
    __sincosf((float)(2*lh*lm)       * PI16, &r0_im, &r0_re);
    __sincosf((float)((2*lh+1)*lm)   * PI16, &r1_im, &r1_re);
    r0_re *= SC; r0_im *= SC; r1_re *= SC; r1_im *= SC;

    // ---- psi' = QFT @ psi via fp32 WMMA ----------------------------------
    v8f a_rl = {};  // Re(psi') rows 0..15   x 16 tokens
    v8f a_rh = {};  // Re(psi') rows 16..31
    v8f a_il = {};  // Im(psi') rows 0..15
    v8f a_ih = {};  // Im(psi') rows 16..31

    #pragma unroll
    for (int kc = 0; kc < 8; ++kc) {
        v2f B   = { Q[kc]*c4, Q[kc]*s4 };
        v2f Arl = { r0_re,  r1_re };
        v2f Ail = { r0_im,  r1_im };
        v2f Arh = { r0_re, -r1_re };   // rows m+16: *(-1)^k; k0 even, k1 odd
        v2f Aih = { r0_im, -r1_im };
        a_rl = __builtin_amdgcn_wmma_f32_16x16x4_f32(false, Arl, false, B, (short)0, a_rl, false, false);
        a_rh = __builtin_amdgcn_wmma_f32_16x16x4_f32(false, Arh, false, B, (short)0, a_rh, false, false);
        a_il = __builtin_amdgcn_wmma_f32_16x16x4_f32(false, Ail, false, B, (short)0, a_il, false, false);
        a_ih = __builtin_amdgcn_wmma_f32_16x16x4_f32(false, Aih, false, B, (short)0, a_ih, false, false);
        // advance coefficients to next chunk: k += 4  (multiply by W)
        float nr = r0_re*w_re - r0_im*w_im;
        float ni = r0_re*w_im + r0_im*w_re;
        r0_re = nr; r0_im = ni;
        nr = r1_re*w_re - r1_im*w_im;
        ni = r1_re*w_im + r1_im*w_re;
        r1_re = nr; r1_im = ni;
    }

    // ---- probs = |psi'|^2; out[i] = sum_j ZS[j,i]*probs[j] ---------------
    // C/D layout: component v in lane L holds row (v + 8*lh) [+16 for hi
    // tile], token column lm.  ZS signs map onto {tile, lh, v bits}.
    float sLo = 0.f, sHi = 0.f, s2w = 0.f, s3w = 0.f, s4w = 0.f;
    #pragma unroll
    for (int v = 0; v < 8; ++v) {
        float pr = a_rl[v]*a_rl[v] + a_il[v]*a_il[v]; // rows  v+8*lh
        float ph = a_rh[v]*a_rh[v] + a_ih[v]*a_ih[v]; // rows 16+v+8*lh
        float q  = pr + ph;
        sLo += pr; sHi += ph;                 // wire0 sign = tile (bit4 of j)
        s2w += (v & 4) ? -q : q;              // wire2 sign = bit2 of j
        s3w += (v & 2) ? -q : q;              // wire3 sign = bit1 of j
        s4w += (v & 1) ? -q : q;              // wire4 sign = bit0 of j
    }
    float e0 = sLo - sHi;
    float t  = sLo + sHi;
    float e1 = lh ? -t : t;                   // wire1 sign = bit3 of j (== lh)

    e0 += __shfl_xor(e0, 16, 32);
    e1 += __shfl_xor(e1, 16, 32);
    float e2 = s2w + __shfl_xor(s2w, 16, 32);
    float e3 = s3w + __shfl_xor(s3w, 16, 32);
    float e4 = s4w + __shfl_xor(s4w, 16, 32);

    if (lh == 0 && tok < ntok) {
        float* op = out + (size_t)tok * WIRES;
        op[0] = e0; op[1] = e1; op[2] = e2; op[3] = e3; op[4] = e4;
    }
}

extern "C" void kernel_launch(void* const* d_in, const int* in_sizes, int n_in,
                              void* d_out, int out_size, void* d_ws, size_t ws_size,
                              hipStream_t stream) {
    const float* x = (const float*)d_in[0];
    // d_in[1] (theta) is mathematically irrelevant: |exp(-i*theta*Z/2)| == 1.
    float* out = (float*)d_out;

    const int ntok  = in_sizes[0] / 512;          // 32*8192 = 262144
    const int tiles = (ntok + 15) / 16;           // 16 tokens per wave
    const int wpb   = 8;                          // 8 waves = 256 threads
    dim3 block(32 * wpb);
    dim3 grid((tiles + wpb - 1) / wpb);
    qft_head_kernel<<<grid, block, 0, stream>>>(x, out, ntok);
}